// FeatureIntegrationModule_12790412607989
// MI455X (gfx1250) — compile-verified
//
#include <hip/hip_runtime.h>

// CDNA5 / gfx1250. wave32. WMMA-based "sum groups of 4 rows" reduction:
//   out(16x16 tile) = sum_{b=0..15} (e_b (x) ones(1x4)) x in[4b..4b+3, 16 cols]
// via chained V_WMMA_F32_16X16X4_F32 (the only f32-input WMMA), accumulator in C/D.
// Memory traffic is identical to the scalar reduction (each input read once,
// each output written once) -> 23.3 TB/s roofline, ~27.5us floor for 640 MiB.
//
// v2: specialize on C so all 32 loads / 8 stores per tile use one per-lane base
// address plus immediate 24-bit byte offsets (span < 512 KB << 8 MB limit),
// killing the per-iteration v_add_nc_u64 address chains and s_wait_xcnt stalls.

typedef __attribute__((ext_vector_type(2))) float v2f;
typedef __attribute__((ext_vector_type(8))) float v8f;

template <int C_CONST>
__global__ __launch_bounds__(256)
void feat_sum4_wmma_kernel(const float* __restrict__ in, float* __restrict__ out,
                           int C_rt, int col_tiles, int total_tiles)
{
    const int C = (C_CONST > 0) ? C_CONST : C_rt;

    // one wave (32 lanes) per 16x16 output tile
    const int wave = blockIdx.x * (blockDim.x >> 5) + (threadIdx.x >> 5);
    if (wave >= total_tiles) return;          // uniform per wave: EXEC stays all-1s for WMMA

    const int lane = threadIdx.x & 31;
    const int half = lane >> 4;               // 0: lanes 0-15, 1: lanes 16-31
    const int nsub = lane & 15;               // column within tile / M within half

    const int row_tile = wave / col_tiles;    // 16 output rows == 64 input rows
    const int col_tile = wave - row_tile * col_tiles;
    const int col      = (col_tile << 4) + nsub;

    // Per-lane base pointers; all per-step displacements become immediates
    // when C_CONST > 0.
    const long long in_r0 = (long long)row_tile * 64 + 2 * half;
    const float* __restrict__ pbase = in + in_r0 * (long long)C + col;

    v8f acc = {};                             // C/D accumulator, 16x16 f32

#pragma unroll
    for (int b = 0; b < 16; ++b) {
        // B-matrix (4x16, f32, 2 VGPRs): lanes hold column `col`; this lane's two
        // VGPR slots are two consecutive input rows of the 4-row slab. The exact
        // K->slot mapping is irrelevant: A is constant along K, so D[b][n] is the
        // plain sum of the 4 values at column n.
        v2f bm;
        bm.x = pbase[(long long)(4 * b) * C];      // row 4b + 2*half
        bm.y = pbase[(long long)(4 * b + 1) * C];  // row 4b + 2*half + 1
        // A-matrix (16x4, f32): A[m][k] = (m == b) ? 1 : 0 for all k.
        // Layout: lanes 0-15 and 16-31 both map M=0..15; both VGPR slots get av.
        const float av = (nsub == b) ? 1.0f : 0.0f;
        v2f am = { av, av };
        // (neg_a, A, neg_b, B, c_mod, C, reuse_a, reuse_b)
        acc = __builtin_amdgcn_wmma_f32_16x16x4_f32(
                  false, am, false, bm, (short)0, acc, false, false);
    }

    // D layout: VGPR v -> M = v (lanes 0-15) or M = v+8 (lanes 16-31), N = nsub.
    const long long orow0 = (long long)row_tile * 16 + 8 * half;
    float* __restrict__ obase = out + orow0 * (long long)C + col;
#pragma unroll
    for (int v = 0; v < 8; ++v) {
        obase[(long long)v * C] = acc[v];
    }
}

__global__ __launch_bounds__(256)
void pids_stride4_kernel(const int* __restrict__ pids, int* __restrict__ out, int chunk)
{
    const int i = blockIdx.x * blockDim.x + threadIdx.x;
    if (i < chunk) out[i] = pids[4 * i];
}

extern "C" void kernel_launch(void* const* d_in, const int* in_sizes, int n_in,
                              void* d_out, int out_size, void* d_ws, size_t ws_size,
                              hipStream_t stream)
{
    const float* features = (const float*)d_in[0];
    const int*   pids     = (const int*)d_in[1];

    const int N     = in_sizes[1];            // 65536
    const int C     = in_sizes[0] / N;        // 2048
    const int chunk = N / 4;                  // 16384

    float* out_feat = (float*)d_out;
    int*   out_pids = (int*)((char*)d_out + (size_t)chunk * (size_t)C * sizeof(float));

    const int row_tiles   = chunk / 16;       // 1024
    const int col_tiles   = C / 16;           // 128
    const int total_tiles = row_tiles * col_tiles;

    const int waves_per_block = 8;            // 256 threads = 8 wave32
    dim3 block(32 * waves_per_block);
    dim3 grid((total_tiles + waves_per_block - 1) / waves_per_block);

    if (C == 2048) {
        feat_sum4_wmma_kernel<2048><<<grid, block, 0, stream>>>(
            features, out_feat, C, col_tiles, total_tiles);
    } else {
        feat_sum4_wmma_kernel<0><<<grid, block, 0, stream>>>(
            features, out_feat, C, col_tiles, total_tiles);
    }

    dim3 pblock(256);
    dim3 pgrid((chunk + 255) / 256);
    pids_stride4_kernel<<<pgrid, pblock, 0, stream>>>(pids, out_pids, chunk);
}